// Conv2DScratch_82025285419642
// MI455X (gfx1250) — compile-verified
//
#include <hip/hip_runtime.h>

// CDNA5 / gfx1250: wave32, V_WMMA_F32_16X16X4_F32 (exact f32 path).
typedef __attribute__((ext_vector_type(2))) float v2f;
typedef __attribute__((ext_vector_type(8))) float v8f;

#define IN_DIM 4096
#define OUT_DIM 4082   // 4096 - 15 + 1
#define KW 15
#define NTILE 256      // ceil(4082/16) tiles per dimension
#define WAVES_PER_BLOCK 8

// Banded-Toeplitz B table in LDS: for each ky (15) and K-step kk (8),
// a 32-lane record of float2 pre-swizzled into the WMMA B operand layout:
//   lane L: h = L>>4 (K half), n = L&15 (output column)
//   .x -> B[c = kk*4 + 2h    ][n],  .y -> B[c = kk*4 + 2h + 1][n]
//   where B[c][n] = w[ky][c-n] if 0 <= c-n < 15 else 0.
#define B_ENTRIES (KW * 8 * 32)

__global__ __launch_bounds__(256) void
conv15_wmma_f32(const float* __restrict__ x,
                const float* __restrict__ w,
                const float* __restrict__ bias,
                float* __restrict__ out)
{
    __shared__ float2 Bs[B_ENTRIES];

    const int tid = threadIdx.x;

    // ---- Build banded B matrices once per workgroup (30 KB LDS) ----
    for (int i = tid; i < B_ENTRIES; i += 256) {
        const int ky  = i >> 8;        // i / 256
        const int rem = i & 255;
        const int kk  = rem >> 5;      // K-step 0..7
        const int L   = rem & 31;      // lane slot
        const int h   = L >> 4;
        const int n   = L & 15;
        const int c   = kk * 4 + 2 * h;
        const int d0  = c - n;
        const int d1  = d0 + 1;
        float bx = (d0 >= 0 && d0 < KW) ? w[ky * KW + d0] : 0.0f;
        float by = (d1 >= 0 && d1 < KW) ? w[ky * KW + d1] : 0.0f;
        Bs[i] = make_float2(bx, by);
    }
    __syncthreads();

    // ---- Per-wave 16x16 output tile ----
    const int lane = tid & 31;
    const int wv   = tid >> 5;
    const int m    = lane & 15;   // A-operand row / C-operand column index
    const int h    = lane >> 4;   // K half (A/B), M half (C/D)

    const int tileX = blockIdx.x * WAVES_PER_BLOCK + wv;  // 0..255
    const int tileY = blockIdx.y;                         // 0..255
    const int row0  = tileY * 16;
    const int col0  = tileX * 16;

    v8f acc = {};

    for (int ky = 0; ky < KW; ++ky) {
        int r = row0 + ky + m;                 // input row this lane feeds
        if (r > IN_DIM - 1) r = IN_DIM - 1;    // clamp (edge tiles only)
        const float*  rowp = x + (size_t)r * IN_DIM;
        const float2* brow = &Bs[(ky * 8) * 32 + lane];

#pragma unroll
        for (int kk = 0; kk < 8; ++kk) {
            int col = col0 + kk * 4 + 2 * h;   // even -> 8B aligned
            if (col > IN_DIM - 2) col = IN_DIM - 2;  // keep float2 in-bounds

            // A operand: lane L supplies A[m, c] / A[m, c+1]
            v2f a = *reinterpret_cast<const v2f*>(rowp + col);

            // B operand from LDS (pre-swizzled)
            float2 bf = brow[kk * 32];
            v2f b;
            b.x = bf.x;
            b.y = bf.y;

            // D = A(16x4) * B(4x16) + C   -- exact f32 MACs
            acc = __builtin_amdgcn_wmma_f32_16x16x4_f32(
                /*neg_a=*/false, a, /*neg_b=*/false, b,
                /*c_mod=*/(short)0, acc, /*reuse_a=*/false, /*reuse_b=*/false);
        }
    }

    // ---- Store: lane L holds N=L&15, VGPR v holds M = v + 8*(L>>4) ----
    const float b0 = bias[0];
    const int n = m;
#pragma unroll
    for (int v = 0; v < 8; ++v) {
        const int rr = row0 + v + 8 * h;
        const int cc = col0 + n;
        if (rr < OUT_DIM && cc < OUT_DIM) {
            out[(size_t)rr * OUT_DIM + cc] = acc[v] + b0;
        }
    }
}

extern "C" void kernel_launch(void* const* d_in, const int* in_sizes, int n_in,
                              void* d_out, int out_size, void* d_ws, size_t ws_size,
                              hipStream_t stream)
{
    (void)in_sizes; (void)n_in; (void)d_ws; (void)ws_size; (void)out_size;
    const float* x    = (const float*)d_in[0];   // 4096*4096 f32
    const float* w    = (const float*)d_in[1];   // 15*15 f32
    const float* bias = (const float*)d_in[2];   // 1 f32

    float* out = (float*)d_out;                  // 4082*4082 f32

    dim3 grid(NTILE / WAVES_PER_BLOCK, NTILE, 1);  // (32, 256)
    conv15_wmma_f32<<<grid, 256, 0, stream>>>(x, w, bias, out);
}